// Attention_70832600646285
// MI455X (gfx1250) — compile-verified
//
#include <hip/hip_runtime.h>
#include <hip/hip_bf16.h>
#include <math.h>

// Problem constants (from reference)
#define B_ 4
#define T_ 512
#define S_ 512
#define IN_ 512
#define D_ 256   // MEM_DIM

typedef __attribute__((ext_vector_type(2))) float v2f;
typedef __attribute__((ext_vector_type(8))) float v8f;

// Native CDNA5 V_TANH_F32 if the toolchain exposes it; OCML fallback.
#if defined(__has_builtin)
#if __has_builtin(__builtin_amdgcn_tanhf)
#define FAST_TANH(x) __builtin_amdgcn_tanhf(x)
#endif
#endif
#ifndef FAST_TANH
#define FAST_TANH(x) tanhf(x)
#endif

// ---------------------------------------------------------------------------
// Generic fp32 WMMA GEMM:  C[M,N] = A[M,K] * W[N,K]^T (+ bias[N])
// One wave per 16x16 output tile, K-loop in steps of 4 using
// V_WMMA_F32_16X16X4_F32.
//
// A-matrix lane layout (32-bit, 16x4): lanes 0-15 hold rows m0..m0+15 with
// K = k0,k0+1 ; lanes 16-31 hold the same rows with K = k0+2,k0+3.
// B-matrix (4x16) mirrors this with columns striped across lanes.
// C/D layout: VGPR j -> M = m0 + j (+8 for upper half-wave), N = lane&15.
// ---------------------------------------------------------------------------
__global__ __launch_bounds__(256) void gemm_nt_kernel(
    const float* __restrict__ A, const float* __restrict__ W,
    const float* __restrict__ bias, float* __restrict__ C,
    int M, int N, int K) {
  int wave = blockIdx.x * (blockDim.x >> 5) + (threadIdx.x >> 5);
  int lane = threadIdx.x & 31;
  int ntn  = N >> 4;
  int m0 = (wave / ntn) << 4;
  int n0 = (wave % ntn) << 4;
  if (m0 >= M) return;                       // wave-uniform; EXEC stays all-1s

  int r  = lane & 15;                        // row (A) / col (B) within tile
  int kk = (lane >> 4) << 1;                 // K sub-offset: 0 or 2
  const float* Arow = A + (size_t)(m0 + r) * K + kk;
  const float* Wcol = W + (size_t)(n0 + r) * K + kk;

  v8f acc = {};
#pragma unroll 4
  for (int k0 = 0; k0 < K; k0 += 4) {
    v2f a = *(const v2f*)(Arow + k0);        // 8B-aligned: kk even, K even
    v2f b = *(const v2f*)(Wcol + k0);
    acc = __builtin_amdgcn_wmma_f32_16x16x4_f32(
        false, a, false, b, (short)0, acc, false, false);
  }

  int half = lane >> 4;
  int col  = n0 + r;
  float bv = bias ? bias[col] : 0.0f;
#pragma unroll
  for (int j = 0; j < 8; ++j)
    C[(size_t)(m0 + j + half * 8) * N + col] = acc[j] + bv;
}

// ---------------------------------------------------------------------------
// Fused score -> masked softmax -> context kernel.
// One block = one (batch b, 16-row t-tile). 256 threads = 8 waves.
//   Phase 1: align[tl,s] = sum_d v[d]*tanh(wq[t0+tl,d] + uh[b,s,d])
//            (wq tile + v in LDS; uh rows broadcast-read from L2;
//             native V_TANH_F32, float2 loads, 2-way accumulator ILP)
//   Phase 2: masked softmax over s (16 threads per row), write align_vectors
//   Phase 3: c[16,256] = P[16,512] * mems[b]  via f32 WMMA (A from LDS)
// ---------------------------------------------------------------------------
__global__ __launch_bounds__(256) void attn_score_ctx_kernel(
    const float* __restrict__ wq,        // [B*T, D]
    const float* __restrict__ uh,        // [B*S, D]
    const float* __restrict__ mems,      // [B, S, D]
    const float* __restrict__ v,         // [D]
    const int*   __restrict__ mem_masks, // [B]
    float* __restrict__ alignOut,        // [B, T, S]
    float* __restrict__ cOut)            // [B*T, D]
{
  __shared__ float sWq[16 * D_];         // 16 KB
  __shared__ float sP[16 * S_];          // 32 KB   scores -> probabilities
  __shared__ float sV[D_];               // 1 KB
  __shared__ float red[256];
  __shared__ float rowStat[16];

  const int tid   = threadIdx.x;
  const int b     = blockIdx.x >> 5;     // T/16 = 32 tiles per batch
  const int t0    = (blockIdx.x & 31) << 4;
  const int len   = mem_masks[b];

  // --- stage wq tile and v into LDS ---
  const float* wqBase = wq + ((size_t)b * T_ + t0) * D_;
  for (int i = tid; i < 16 * D_; i += 256) sWq[i] = wqBase[i];
  for (int i = tid; i < D_;      i += 256) sV[i]  = v[i];
  __syncthreads();

  // --- Phase 1: additive-attention scores (the 268M-tanh hot loop) ---
  {
    const float* uhB = uh + (size_t)b * S_ * D_;
    const int tl = tid & 15;                    // fixed row per thread
    const v2f* wqRow = (const v2f*)(sWq + tl * D_);
    const v2f* vv    = (const v2f*)sV;
#pragma unroll 1
    for (int i = 0; i < (16 * S_) / 256; ++i) { // 32 (tl,s) pairs / thread
      const int s = (tid >> 4) + i * 16;
      const v2f* uhRow = (const v2f*)(uhB + (size_t)s * D_);
      float a0 = 0.0f, a1 = 0.0f;               // 2-way ILP for TRANS pipe
#pragma unroll 4
      for (int d = 0; d < D_ / 2; ++d) {
        v2f u = uhRow[d];
        v2f w = wqRow[d];
        v2f vd = vv[d];
        a0 = fmaf(vd.x, FAST_TANH(w.x + u.x), a0);
        a1 = fmaf(vd.y, FAST_TANH(w.y + u.y), a1);
      }
      sP[tl * S_ + s] = a0 + a1;
    }
  }
  __syncthreads();

  // --- Phase 2: masked softmax over s, per row (16 threads / row) ---
  const int row = tid >> 4;
  const int sub = tid & 15;
  float mx = -INFINITY;
  for (int s = sub; s < S_; s += 16)
    if (s < len) mx = fmaxf(mx, sP[row * S_ + s]);
  red[tid] = mx;
  __syncthreads();
  if (sub == 0) {
    float m = red[row * 16];
    for (int j = 1; j < 16; ++j) m = fmaxf(m, red[row * 16 + j]);
    rowStat[row] = m;
  }
  __syncthreads();
  const float rm = rowStat[row];
  float sum = 0.0f;
  for (int s = sub; s < S_; s += 16) {
    float e = (s < len) ? __expf(sP[row * S_ + s] - rm) : 0.0f;
    sP[row * S_ + s] = e;
    sum += e;
  }
  red[tid] = sum;
  __syncthreads();
  if (sub == 0) {
    float t = 0.0f;
    for (int j = 0; j < 16; ++j) t += red[row * 16 + j];
    rowStat[row] = t;
  }
  __syncthreads();
  const float inv = 1.0f / rowStat[row];
  float* aOut = alignOut + ((size_t)b * T_ + t0 + row) * S_;
  for (int s = sub; s < S_; s += 16) {
    float p = sP[row * S_ + s] * inv;
    sP[row * S_ + s] = p;
    aOut[s] = p;                               // masked slots are exactly 0
  }
  __syncthreads();

  // --- Phase 3: context c = P[16,512] * mems[b][512,256] via f32 WMMA ---
  {
    const int wv   = tid >> 5;
    const int lane = tid & 31;
    const int r    = lane & 15;
    const int kk   = (lane >> 4) << 1;
    const float* memB = mems + (size_t)b * S_ * D_;
    float* cBase = cOut + ((size_t)b * T_ + t0) * D_;
    for (int nt = wv; nt < D_ / 16; nt += 8) { // 16 n-tiles over 8 waves
      const int n0 = nt << 4;
      v8f acc = {};
#pragma unroll 4
      for (int k0 = 0; k0 < S_; k0 += 4) {
        const int kb = k0 + kk;
        v2f a = *(const v2f*)(sP + r * S_ + kb);        // LDS, 8B aligned
        v2f bm;
        bm.x = memB[(size_t)kb * D_ + n0 + r];          // B: col-striped lanes
        bm.y = memB[(size_t)(kb + 1) * D_ + n0 + r];
        acc = __builtin_amdgcn_wmma_f32_16x16x4_f32(
            false, a, false, bm, (short)0, acc, false, false);
      }
      const int half = lane >> 4;
      const int col  = n0 + r;
#pragma unroll
      for (int j = 0; j < 8; ++j)
        cBase[(size_t)(j + half * 8) * D_ + col] = acc[j];
    }
  }
}

// ---------------------------------------------------------------------------
// Output projection: attn_h = concat(c, inputs) * Wout^T + bout
// Implemented as a split-K WMMA GEMM (K = 256 from c, then 512 from inputs),
// so the concat is never materialized. M = B*T = 2048, N = 512, K = 768.
// ---------------------------------------------------------------------------
__global__ __launch_bounds__(256) void outproj_kernel(
    const float* __restrict__ Cctx,   // [B*T, 256]
    const float* __restrict__ X,      // [B*T, 512]
    const float* __restrict__ W,      // [512, 768]
    const float* __restrict__ bias,   // [512]
    float* __restrict__ Out)          // [B*T, 512]
{
  const int wave = blockIdx.x * 8 + (threadIdx.x >> 5);
  const int lane = threadIdx.x & 31;
  const int m0 = (wave >> 5) << 4;      // 32 n-tiles (N=512)
  const int n0 = (wave & 31) << 4;
  const int r  = lane & 15;
  const int kk = (lane >> 4) << 1;

  const float* Ac = Cctx + (size_t)(m0 + r) * D_ + kk;
  const float* Ax = X    + (size_t)(m0 + r) * IN_ + kk;
  const float* Wc = W    + (size_t)(n0 + r) * (D_ + IN_) + kk;

  v8f acc = {};
#pragma unroll 4
  for (int k0 = 0; k0 < D_; k0 += 4) {       // K segment 1: context
    v2f a = *(const v2f*)(Ac + k0);
    v2f w = *(const v2f*)(Wc + k0);
    acc = __builtin_amdgcn_wmma_f32_16x16x4_f32(
        false, a, false, w, (short)0, acc, false, false);
  }
#pragma unroll 4
  for (int k0 = 0; k0 < IN_; k0 += 4) {      // K segment 2: inputs
    v2f a = *(const v2f*)(Ax + k0);
    v2f w = *(const v2f*)(Wc + D_ + k0);
    acc = __builtin_amdgcn_wmma_f32_16x16x4_f32(
        false, a, false, w, (short)0, acc, false, false);
  }

  const int half = lane >> 4;
  const int col  = n0 + r;
  const float bv = bias[col];
#pragma unroll
  for (int j = 0; j < 8; ++j)
    Out[(size_t)(m0 + j + half * 8) * IN_ + col] = acc[j] + bv;
}

// ---------------------------------------------------------------------------
extern "C" void kernel_launch(void* const* d_in, const int* in_sizes, int n_in,
                              void* d_out, int out_size, void* d_ws, size_t ws_size,
                              hipStream_t stream) {
  const float* inputs    = (const float*)d_in[0];  // [B,T,IN]
  const float* mems      = (const float*)d_in[1];  // [B,S,D]
  const int*   mem_masks = (const int*)  d_in[2];  // [B]
  const float* Wq        = (const float*)d_in[3];  // [D,IN]
  const float* Wc        = (const float*)d_in[4];  // [D,D]
  const float* bc        = (const float*)d_in[5];  // [D]
  const float* v         = (const float*)d_in[6];  // [D]
  const float* Wout      = (const float*)d_in[7];  // [IN, D+IN]
  const float* bout      = (const float*)d_in[8];  // [IN]

  float* attn  = (float*)d_out;                        // [B,T,IN]
  float* align = attn + (size_t)B_ * T_ * IN_;         // [B,T,S]

  float* ws  = (float*)d_ws;
  float* wq  = ws;                                     // [B*T, D]
  float* uh  = wq + (size_t)B_ * T_ * D_;              // [B*S, D]
  float* ctx = uh + (size_t)B_ * S_ * D_;              // [B*T, D]

  const int M = B_ * T_;                               // 2048 (== B*S too)

  // 1) wq = inputs * Wq^T          : 2048 tiles -> 256 blocks of 8 waves
  gemm_nt_kernel<<<(M / 16) * (D_ / 16) / 8, 256, 0, stream>>>(
      inputs, Wq, nullptr, wq, M, D_, IN_);

  // 2) uh = mems * Wc^T + bc
  gemm_nt_kernel<<<(M / 16) * (D_ / 16) / 8, 256, 0, stream>>>(
      mems, Wc, bc, uh, M, D_, D_);

  // 3) fused tanh-score + masked softmax + context (one block per 16 t-rows)
  attn_score_ctx_kernel<<<B_ * (T_ / 16), 256, 0, stream>>>(
      wq, uh, mems, v, mem_masks, align, ctx);

  // 4) attn_h = concat(ctx, inputs) * Wout^T + bout : 4096 tiles -> 512 blocks
  outproj_kernel<<<(M / 16) * (IN_ / 16) / 8, 256, 0, stream>>>(
      ctx, inputs, Wout, bout, attn);
}